// ProteinGCN_4123168604928
// MI455X (gfx1250) — compile-verified
//
#include <hip/hip_runtime.h>
#include <hip/hip_bf16.h>

typedef __attribute__((ext_vector_type(2))) float v2f;
typedef __attribute__((ext_vector_type(8))) float v8f;

// ---------------------------------------------------------------------------
// Utility kernels
// ---------------------------------------------------------------------------
__global__ void fill_kernel(float* __restrict__ p, float v, long long n) {
    long long i = (long long)blockIdx.x * blockDim.x + threadIdx.x;
    if (i < n) p[i] = v;
}

__global__ void degree_kernel(const long long* __restrict__ dst,
                              float* __restrict__ deg, long long ne) {
    long long i = (long long)blockIdx.x * blockDim.x + threadIdx.x;
    if (i < ne) atomicAdd(deg + dst[i], 1.0f);
}

__global__ void rsqrt_kernel(float* __restrict__ d, int n) {
    int i = blockIdx.x * blockDim.x + threadIdx.x;
    if (i < n) d[i] = rsqrtf(d[i]);
}

// ---------------------------------------------------------------------------
// fp32 WMMA GEMM: C[M x NDIM] = X[M x KDIM] @ W[KDIM x NDIM]
// One wave per 16x16 C tile, looping K in steps of 4 using
// V_WMMA_F32_16X16X4_F32. Block = (NDIM/16) waves covering one 16-row strip.
//
// A 16x4 layout:  lanes 0-15 hold M=0..15; VGPR0={K=0 | K=2}, VGPR1={K=1 | K=3}
// B 4x16 layout:  lanes 0-15 cols N=0..15; VGPR0={K=0 | K=2}, VGPR1={K=1 | K=3}
// C/D layout:     VGPR r: lanes 0-15 -> (M=r, N=lane); lanes 16-31 -> (M=r+8)
// ---------------------------------------------------------------------------
template <int KDIM, int NDIM>
__global__ __launch_bounds__((NDIM / 16) * 32)
void gemm_wmma_f32(const float* __restrict__ X, const float* __restrict__ W,
                   float* __restrict__ C) {
    const int lane  = threadIdx.x & 31;
    const int wave  = threadIdx.x >> 5;
    const int m0    = blockIdx.x * 16;
    const int n0    = wave * 16;
    const int lrow  = lane & 15;         // A row / B col within tile
    const int khalf = (lane >> 4) << 1;  // 0 for lanes 0-15, 2 for lanes 16-31

    const float* __restrict__ Xrow = X + (long long)(m0 + lrow) * KDIM + khalf;

    v8f acc = {};
#pragma unroll
    for (int k = 0; k < KDIM; k += 4) {
        // A fragment: two adjacent K elements -> 8-byte aligned vector load
        v2f a = *(const v2f*)(Xrow + k);
        // B fragment: two consecutive K rows of W, same column
        v2f b;
        b.x = W[(long long)(k + khalf + 0) * NDIM + n0 + lrow];
        b.y = W[(long long)(k + khalf + 1) * NDIM + n0 + lrow];
        acc = __builtin_amdgcn_wmma_f32_16x16x4_f32(
            /*neg_a=*/false, a, /*neg_b=*/false, b,
            /*c_mod=*/(short)0, acc, /*reuse_a=*/false, /*reuse_b=*/false);
    }

    const int cm = m0 + ((lane >> 4) << 3);  // +8 for high half-wave
    const int cn = n0 + lrow;
#pragma unroll
    for (int r = 0; r < 8; ++r)
        C[(long long)(cm + r) * NDIM + cn] = acc[r];
}

// ---------------------------------------------------------------------------
// Edge scatter: one wave per edge. acc[dst] += xw[src] * dinv[src]*dinv[dst]
// H = feature width (128 or 64); VEC = H/32 floats per lane.
// Both xw (<=51MB) and acc (<=51MB) are L2-resident on MI455X (192MB L2),
// so the random gather + f32 atomics stay on-chip.
// ---------------------------------------------------------------------------
template <int H, int VEC>
__global__ __launch_bounds__(256)
void scatter_edges(const float* __restrict__ xw, const long long* __restrict__ src,
                   const long long* __restrict__ dst, const float* __restrict__ dinv,
                   float* __restrict__ acc, long long nedges) {
    const int lane = threadIdx.x & 31;
    const long long e =
        (long long)blockIdx.x * (blockDim.x >> 5) + (threadIdx.x >> 5);
    if (e >= nedges) return;

    if (lane == 0) {  // streaming prefetch of upcoming edge indices
        __builtin_prefetch(src + e + 4096, 0, 1);
        __builtin_prefetch(dst + e + 4096, 0, 1);
    }

    const long long s = src[e];
    const long long d = dst[e];
    const float norm = dinv[s] * dinv[d];

    const float* __restrict__ in = xw + s * H + lane * VEC;
    float* __restrict__ out = acc + d * H + lane * VEC;

    float vals[VEC];
#pragma unroll
    for (int j = 0; j < VEC; ++j) vals[j] = in[j];  // contiguous -> b128/b64 load
#pragma unroll
    for (int j = 0; j < VEC; ++j) atomicAdd(out + j, vals[j] * norm);
}

// ---------------------------------------------------------------------------
// Finalize: out = relu(acc + xw * dinv^2 (self loop) + bias)
// ---------------------------------------------------------------------------
template <int H>
__global__ void finalize_relu(const float* __restrict__ acc,
                              const float* __restrict__ xw,
                              const float* __restrict__ dinv,
                              const float* __restrict__ bias,
                              float* __restrict__ out, int n) {
    long long idx = (long long)blockIdx.x * blockDim.x + threadIdx.x;
    if (idx >= (long long)n * H) return;
    const int node = (int)(idx >> (H == 128 ? 7 : 6));
    const int j = (int)(idx & (H - 1));
    const float di = dinv[node];
    float v = acc[idx] + xw[idx] * di * di + bias[j];
    out[idx] = v > 0.0f ? v : 0.0f;
}

// ---------------------------------------------------------------------------
// FC head: out[i] = dot(h[i, 0:64], Wfc) + bfc ; one wave per node (wave32)
// ---------------------------------------------------------------------------
__global__ __launch_bounds__(256)
void fc_kernel(const float* __restrict__ h, const float* __restrict__ Wfc,
               const float* __restrict__ bfc, float* __restrict__ out, int n) {
    const int lane = threadIdx.x & 31;
    const int node = blockIdx.x * (blockDim.x >> 5) + (threadIdx.x >> 5);
    if (node >= n) return;
    const float* __restrict__ row = h + (long long)node * 64 + lane * 2;
    float sum = row[0] * Wfc[lane * 2] + row[1] * Wfc[lane * 2 + 1];
#pragma unroll
    for (int off = 16; off > 0; off >>= 1) sum += __shfl_xor(sum, off, 32);
    if (lane == 0) out[node] = sum + bfc[0];
}

// ---------------------------------------------------------------------------
// Launch
// ---------------------------------------------------------------------------
extern "C" void kernel_launch(void* const* d_in, const int* in_sizes, int n_in,
                              void* d_out, int out_size, void* d_ws, size_t ws_size,
                              hipStream_t stream) {
    constexpr int IN_DIM = 64, H1 = 128, H2 = 64;

    const float*     x   = (const float*)d_in[0];
    const long long* ei  = (const long long*)d_in[1];
    const float*     W1  = (const float*)d_in[2];
    const float*     b1  = (const float*)d_in[3];
    const float*     W2  = (const float*)d_in[4];
    const float*     b2  = (const float*)d_in[5];
    const float*     Wfc = (const float*)d_in[6];
    const float*     bfc = (const float*)d_in[7];
    float*           out = (float*)d_out;

    const int N = in_sizes[0] / IN_DIM;      // 100000
    const long long E = in_sizes[1] / 2;     // 1600000
    const long long* src = ei;
    const long long* dst = ei + E;

    // Workspace: dinv[N] | bufA[N*128] | bufB[N*128]   (~103 MB)
    float* dinv = (float*)d_ws;
    float* bufA = dinv + N;
    float* bufB = bufA + (size_t)N * H1;

    const int mtiles = N / 16;  // 6250 (N divisible by 16)

    // --- degrees -> dinv -------------------------------------------------
    fill_kernel<<<(N + 255) / 256, 256, 0, stream>>>(dinv, 1.0f, N);
    degree_kernel<<<(int)((E + 255) / 256), 256, 0, stream>>>(dst, dinv, E);
    rsqrt_kernel<<<(N + 255) / 256, 256, 0, stream>>>(dinv, N);

    // --- layer 1: xw1 = x @ W1 ------------------------------------------
    gemm_wmma_f32<IN_DIM, H1><<<mtiles, (H1 / 16) * 32, 0, stream>>>(x, W1, bufA);

    fill_kernel<<<(int)(((long long)N * H1 + 255) / 256), 256, 0, stream>>>(
        bufB, 0.0f, (long long)N * H1);
    scatter_edges<H1, H1 / 32><<<(int)((E + 7) / 8), 256, 0, stream>>>(
        bufA, src, dst, dinv, bufB, E);
    finalize_relu<H1><<<(int)(((long long)N * H1 + 255) / 256), 256, 0, stream>>>(
        bufB, bufA, dinv, b1, bufB, N);

    // --- layer 2: xw2 = h1 @ W2 -----------------------------------------
    gemm_wmma_f32<H1, H2><<<mtiles, (H2 / 16) * 32, 0, stream>>>(bufB, W2, bufA);

    fill_kernel<<<(int)(((long long)N * H2 + 255) / 256), 256, 0, stream>>>(
        bufB, 0.0f, (long long)N * H2);
    scatter_edges<H2, H2 / 32><<<(int)((E + 7) / 8), 256, 0, stream>>>(
        bufA, src, dst, dinv, bufB, E);
    finalize_relu<H2><<<(int)(((long long)N * H2 + 255) / 256), 256, 0, stream>>>(
        bufB, bufA, dinv, b2, bufB, N);

    // --- FC head ---------------------------------------------------------
    fc_kernel<<<(N + 7) / 8, 256, 0, stream>>>(bufB, Wfc, bfc, out, N);
}